// Clip_autoencoder_49134425866737
// MI455X (gfx1250) — compile-verified
//
#include <hip/hip_runtime.h>
#include <hip/hip_bf16.h>
#include <cstdint>
#include <cstddef>

typedef __attribute__((ext_vector_type(2))) float v2f;
typedef __attribute__((ext_vector_type(8))) float v8f;

#define BATCH 2048

// conv kernel flags
#define CF_RELU_IN    2
#define CF_RELU_OUT   4
#define CF_OUT_BMAJOR 8

// ---------------------------------------------------------------------------
// Repack weights into [tap][ci][co] (co contiguous) so the GEMM kernel gets
// fully coalesced A-fragment loads and identical code for conv and convT.
//   conv :  w[co][ci][tap]   (torch Conv2d  [Cout][Cin][k][k])
//   convT:  w[ci][co][tap]   (torch ConvT2d [Cin][Cout][k][k])
// ---------------------------------------------------------------------------
__global__ __launch_bounds__(256) void repack_w_kernel(
    const float* __restrict__ w, float* __restrict__ wr,
    int Cin, int Cout, int KK, int tposed, int total)
{
  const int idx = blockIdx.x * 256 + threadIdx.x;
  if (idx >= total) return;
  const int co  = idx % Cout;
  const int t1  = idx / Cout;
  const int ci  = t1 % Cin;
  const int tap = t1 / Cin;
  const size_t src = tposed ? ((size_t)ci * Cout + co) * KK + tap
                            : ((size_t)co * Cin + ci) * KK + tap;
  wr[idx] = w[src];  // dst = ((tap*Cin)+ci)*Cout + co == idx
}

// ---------------------------------------------------------------------------
// Implicit-GEMM conv using V_WMMA_F32_16X16X4_F32 with 2x2 register blocking.
// Activations layout: [pix][C][BATCH] (b contiguous).
// Weights: repacked [tap][ci][co].
// Each wave computes a 32(co) x 32(b) tile (4 accumulators) for one output
// pixel: per K=4 step -> 8 dword loads, 4 WMMAs (A/B fragments reused 2x).
// Block = 256 threads = 8 waves -> co 32 x b 256 per block.
// ---------------------------------------------------------------------------
__global__ __launch_bounds__(256) void conv_wmma_kernel(
    const float* __restrict__ in, float* __restrict__ out,
    const float* __restrict__ wr, const float* __restrict__ bias,
    const float* __restrict__ res,
    int Cin, int Cout, int K, int stride, int pad,
    int inH, int outH, int tposed, int flags)
{
  const int lane = threadIdx.x & 31;
  const int wv   = threadIdx.x >> 5;
  const int half = lane >> 4;       // 0: lanes 0-15, 1: lanes 16-31
  const int l    = lane & 15;
  const int bBase  = blockIdx.x * 256 + wv * 32;
  const int coBase = blockIdx.y * 32;
  const int opix = blockIdx.z;
  const int oy = opix / outH;
  const int ox = opix - oy * outH;
  const bool reluIn = (flags & CF_RELU_IN) != 0;

  v8f a00 = {0.f,0.f,0.f,0.f,0.f,0.f,0.f,0.f};
  v8f a01 = a00, a10 = a00, a11 = a00;

  for (int ky = 0; ky < K; ++ky) {
    int iy;
    if (tposed) {
      int t = oy + pad - ky;
      if (t < 0 || (t % stride) != 0) continue;
      iy = t / stride;
      if (iy >= inH) continue;
    } else {
      iy = oy * stride + ky - pad;
      if (iy < 0 || iy >= inH) continue;
    }
    for (int kx = 0; kx < K; ++kx) {
      int ix;
      if (tposed) {
        int t = ox + pad - kx;
        if (t < 0 || (t % stride) != 0) continue;
        ix = t / stride;
        if (ix >= inH) continue;
      } else {
        ix = ox * stride + kx - pad;
        if (ix < 0 || ix >= inH) continue;
      }
      const int tap = ky * K + kx;
      // B: rows ci+2*half+{0,1}, cols bBase+l (+16 for second b tile)
      const float* bp = in + ((size_t)(iy * inH + ix) * Cin + 2 * half) * BATCH
                           + (size_t)bBase + l;
      // A (repacked): [tap][ci][co]; rows ci+2*half+{0,1}, co coBase+l (+16)
      const float* ap = wr + ((size_t)tap * Cin + 2 * half) * Cout
                           + (size_t)coBase + l;
      for (int ci = 0; ci < Cin; ci += 4) {
        v2f av0, av1, bv0, bv1;
        av0.x = ap[0];        av0.y = ap[Cout];        // co tile 0
        av1.x = ap[16];       av1.y = ap[Cout + 16];   // co tile 1
        float b00 = bp[0],  b01 = bp[BATCH];           // b tile 0
        float b10 = bp[16], b11 = bp[BATCH + 16];      // b tile 1
        if (reluIn) {
          b00 = fmaxf(b00, 0.f); b01 = fmaxf(b01, 0.f);
          b10 = fmaxf(b10, 0.f); b11 = fmaxf(b11, 0.f);
        }
        bv0.x = b00; bv0.y = b01;
        bv1.x = b10; bv1.y = b11;
        a00 = __builtin_amdgcn_wmma_f32_16x16x4_f32(false, av0, false, bv0,
                                                    (short)0, a00, false, false);
        a01 = __builtin_amdgcn_wmma_f32_16x16x4_f32(false, av0, false, bv1,
                                                    (short)0, a01, false, false);
        a10 = __builtin_amdgcn_wmma_f32_16x16x4_f32(false, av1, false, bv0,
                                                    (short)0, a10, false, false);
        a11 = __builtin_amdgcn_wmma_f32_16x16x4_f32(false, av1, false, bv1,
                                                    (short)0, a11, false, false);
        ap += (size_t)Cout * 4;
        bp += (size_t)BATCH * 4;
      }
    }
  }

  // Epilogue: D element (m, n): vgpr = m%8, lanes (m<8?0:16)+n
  #pragma unroll
  for (int ct = 0; ct < 2; ++ct) {
    #pragma unroll
    for (int bt = 0; bt < 2; ++bt) {
      const v8f& acc = ct == 0 ? (bt == 0 ? a00 : a01) : (bt == 0 ? a10 : a11);
      #pragma unroll
      for (int v = 0; v < 8; ++v) {
        const int m  = v + 8 * half;
        const int co = coBase + 16 * ct + m;
        const int n  = bBase + 16 * bt + l;
        float val = acc[v];
        if (bias) val += bias[co];
        if (res)  val += res[((size_t)opix * Cout + co) * BATCH + n];
        if (flags & CF_RELU_OUT) val = fmaxf(val, 0.f);
        if (flags & CF_OUT_BMAJOR) out[(size_t)n * Cout + co] = val;
        else out[((size_t)opix * Cout + co) * BATCH + n] = val;
      }
    }
  }
}

// ---------------------------------------------------------------------------
// Transpose x[2048][512] -> y[512][2048]
// ---------------------------------------------------------------------------
__global__ __launch_bounds__(256) void transpose_x(
    const float* __restrict__ x, float* __restrict__ y)
{
  __shared__ float tile[32][33];
  const int c0 = blockIdx.x * 32;
  const int b0 = blockIdx.y * 32;
  const int tx = threadIdx.x & 31;
  const int ty = threadIdx.x >> 5;  // 0..7
  #pragma unroll
  for (int r = ty; r < 32; r += 8)
    tile[r][tx] = x[(size_t)(b0 + r) * 512 + (c0 + tx)];
  __syncthreads();
  #pragma unroll
  for (int r = ty; r < 32; r += 8)
    y[(size_t)(c0 + r) * BATCH + (b0 + tx)] = tile[tx][r];
}

// ---------------------------------------------------------------------------
// ||codebook[e]||^2 for e in [0,1024)
// ---------------------------------------------------------------------------
__global__ __launch_bounds__(256) void cnorm_kernel(
    const float* __restrict__ cb, float* __restrict__ cn)
{
  const int e = blockIdx.x * 256 + threadIdx.x;
  if (e < 1024) {
    float s = 0.f;
    #pragma unroll 8
    for (int k = 0; k < 128; ++k) {
      float v = cb[(size_t)e * 128 + k];
      s = fmaf(v, v, s);
    }
    cn[e] = s;
  }
}

__global__ void zero_kernel(float* __restrict__ p) { *p = 0.f; }

__global__ void loss_final_kernel(const float* __restrict__ acc,
                                  float* __restrict__ dst)
{
  // loss = (1 + 0.25) * sum((zq-zp)^2) / (2048*25*128)
  *dst = *acc * (1.25f / 6553600.f);
}

// ---------------------------------------------------------------------------
// Vector quantizer: one thread per (pix, b). z layout [25][128][2048].
// d = ||z||^2 + ||c||^2 - 2 z.c  -> argmin over (cn - 2*dot).
// Writes straight-through zq in place and accumulates sum((c - zv)^2).
// ---------------------------------------------------------------------------
#define VQ_TILE 64
__global__ __launch_bounds__(256) void vq_kernel(
    const float* __restrict__ z, const float* __restrict__ cb,
    const float* __restrict__ cn, float* __restrict__ zq,
    float* __restrict__ lossAcc)
{
  __shared__ float cbs[VQ_TILE * 128];
  __shared__ float cns[VQ_TILE];
  __shared__ float red[256];
  const int t = blockIdx.x * 256 + threadIdx.x;
  const int b = t & (BATCH - 1);
  const int pix = t >> 11;

  float zv[128];
  #pragma unroll
  for (int k = 0; k < 128; ++k)
    zv[k] = z[(size_t)(pix * 128 + k) * BATCH + b];

  float best = 3.4e38f;
  int bidx = 0;
  for (int e0 = 0; e0 < 1024; e0 += VQ_TILE) {
    __syncthreads();
    for (int i = threadIdx.x; i < VQ_TILE * 128; i += 256)
      cbs[i] = cb[(size_t)e0 * 128 + i];
    if (threadIdx.x < VQ_TILE) cns[threadIdx.x] = cn[e0 + threadIdx.x];
    __syncthreads();
    for (int e = 0; e < VQ_TILE; ++e) {
      const float* cr = &cbs[e * 128];  // broadcast reads: conflict-free
      float dot = 0.f;
      #pragma unroll
      for (int k = 0; k < 128; ++k) dot = fmaf(zv[k], cr[k], dot);
      const float sc = cns[e] - 2.f * dot;
      if (sc < best) { best = sc; bidx = e0 + e; }
    }
  }

  float lsum = 0.f;
  const float* cbest = cb + (size_t)bidx * 128;
  #pragma unroll
  for (int k = 0; k < 128; ++k) {
    const float d = cbest[k] - zv[k];
    zq[(size_t)(pix * 128 + k) * BATCH + b] = zv[k] + d;  // straight-through
    lsum += d * d;
  }
  red[threadIdx.x] = lsum;
  __syncthreads();
  for (int s = 128; s > 0; s >>= 1) {
    if (threadIdx.x < (unsigned)s) red[threadIdx.x] += red[threadIdx.x + s];
    __syncthreads();
  }
  if (threadIdx.x == 0) atomicAdd(lossAcc, red[0]);
}

// ---------------------------------------------------------------------------
static inline void launch_conv(hipStream_t stream,
                               const float* in, float* out,
                               const float* w, float* wrep,
                               const float* bias, const float* res,
                               int Cin, int Cout, int K, int stride, int pad,
                               int inH, int outH, int tposed, int flags)
{
  const int total = Cin * Cout * K * K;
  repack_w_kernel<<<dim3((total + 255) / 256), dim3(256), 0, stream>>>(
      w, wrep, Cin, Cout, K * K, tposed, total);
  conv_wmma_kernel<<<dim3(8, Cout / 32, outH * outH), dim3(256), 0, stream>>>(
      in, out, wrep, bias, res, Cin, Cout, K, stride, pad, inH, outH,
      tposed, flags);
}

extern "C" void kernel_launch(void* const* d_in, const int* in_sizes, int n_in,
                              void* d_out, int out_size, void* d_ws, size_t ws_size,
                              hipStream_t stream)
{
  const float* x       = (const float*)d_in[0];
  const float* enc_w1  = (const float*)d_in[1];
  const float* enc_b1  = (const float*)d_in[2];
  const float* enc_w2  = (const float*)d_in[3];
  const float* enc_b2  = (const float*)d_in[4];
  const float* enc_rw1 = (const float*)d_in[5];   // [3][256][128][3][3]
  const float* enc_rw2 = (const float*)d_in[6];   // [3][128][256][1][1]
  const float* pre_w   = (const float*)d_in[7];
  const float* pre_b   = (const float*)d_in[8];
  const float* cb      = (const float*)d_in[9];   // [1024][128]
  const float* post_w  = (const float*)d_in[10];
  const float* post_b  = (const float*)d_in[11];
  const float* dec_rw1 = (const float*)d_in[12];  // [3][128][256][3][3]
  const float* dec_rw2 = (const float*)d_in[13];  // [3][256][128][1][1]
  const float* f1_w    = (const float*)d_in[14];
  const float* f1_b    = (const float*)d_in[15];
  const float* f2_w    = (const float*)d_in[16];
  const float* f2_b    = (const float*)d_in[17];
  float* out = (float*)d_out;

  float* ws = (float*)d_ws;
  float* S1 = ws;                             //  512*2048      (x^T)
  float* S2 = S1 + (size_t)512 * 2048;        //  9*384*2048
  float* S3 = S2 + (size_t)9 * 384 * 2048;    // 25*256*2048
  float* S4 = S3 + (size_t)25 * 256 * 2048;   // 25*128*2048
  float* CN = S4 + (size_t)25 * 128 * 2048;   // 1024
  float* LS = CN + 1024;                      // 1
  float* WR = LS + 1;                         // 512*384*9 repacked weights

  const dim3 blk(256);

  // ---- Encoder ----
  transpose_x<<<dim3(16, 64), blk, 0, stream>>>(x, S1);
  // convT 512->384 k3 s2 p0 : 1x1 -> 3x3, relu
  launch_conv(stream, S1, S2, enc_w1, WR, enc_b1, nullptr,
              512, 384, 3, 2, 0, 1, 3, 1, CF_RELU_OUT);
  // convT 384->256 k3 s2 p1 : 3x3 -> 5x5, relu  (h)
  launch_conv(stream, S2, S3, enc_w2, WR, enc_b2, nullptr,
              384, 256, 3, 2, 1, 3, 5, 1, CF_RELU_OUT);
  // inverse residual stack
  for (int i = 0; i < 3; ++i) {
    launch_conv(stream, S3, S4, enc_rw1 + (size_t)i * 256 * 128 * 9, WR,
                nullptr, nullptr, 256, 128, 3, 1, 1, 5, 5, 1, CF_RELU_OUT);
    launch_conv(stream, S4, S3, enc_rw2 + (size_t)i * 128 * 256, WR,
                nullptr, S3, 128, 256, 1, 1, 0, 5, 5, 1, CF_RELU_OUT);
  }
  // pre conv 256->128 k3 p1 (z)
  launch_conv(stream, S3, S4, pre_w, WR, pre_b, nullptr,
              256, 128, 3, 1, 1, 5, 5, 0, 0);

  // ---- Vector quantizer ----
  cnorm_kernel<<<dim3(4), blk, 0, stream>>>(cb, CN);
  zero_kernel<<<dim3(1), dim3(1), 0, stream>>>(LS);
  vq_kernel<<<dim3(200), blk, 0, stream>>>(S4, cb, CN, S4, LS);

  // ---- Decoder ----
  // post conv 128->256 k3 p1 (no relu)
  launch_conv(stream, S4, S3, post_w, WR, post_b, nullptr,
              128, 256, 3, 1, 1, 5, 5, 0, 0);
  // residual stack (h may be negative -> relu_in on first conv)
  for (int i = 0; i < 3; ++i) {
    launch_conv(stream, S3, S2, dec_rw1 + (size_t)i * 128 * 256 * 9, WR,
                nullptr, nullptr, 256, 128, 3, 1, 1, 5, 5, 0,
                CF_RELU_IN | CF_RELU_OUT);
    launch_conv(stream, S2, S3, dec_rw2 + (size_t)i * 256 * 128, WR,
                nullptr, S3, 128, 256, 1, 1, 0, 5, 5, 0, CF_RELU_OUT);
  }
  // f1 conv 256->384 k3 s2 p1 : 5x5 -> 3x3, relu
  launch_conv(stream, S3, S2, f1_w, WR, f1_b, nullptr,
              256, 384, 3, 2, 1, 5, 3, 0, CF_RELU_OUT);
  // f2 conv 384->512 k3 s2 p0 : 3x3 -> 1x1 -> d_out[b][co]
  launch_conv(stream, S2, out, f2_w, WR, f2_b, nullptr,
              384, 512, 3, 2, 0, 3, 1, 0, CF_OUT_BMAJOR);

  // loss scalar at d_out[2048*512]
  loss_final_kernel<<<dim3(1), dim3(1), 0, stream>>>(LS, out + (size_t)2048 * 512);
}